// LSSLKernel_84361747628468
// MI455X (gfx1250) — compile-verified
//
#include <hip/hip_runtime.h>
#include <hip/hip_bf16.h>

// LSSL / S4-style SSM convolution-kernel generation for MI455X (gfx1250).
//
//   K[h,l] = Re( sum_n w[h,n] * a[h,n]^l ),  w = C*b_bar, a = a_bar (bilinear)
//
// Decompose l = 16*q + r:  a^l = (a^16)^q * a^r
//   => per h:  K_tile[q,r] = Re( G @ S ),  G = (a^16)^q  (128x64 complex),
//                                          S = w * a^r   (64x16 complex)
//   => two real f32 GEMMs (M=128,K=64,N=16) on V_WMMA_F32_16X16X4_F32.
//
// One block (8 wave32) per channel h; wave w owns q-tile [16w,16w+16).
// A-fragment generation: since lane exponent q = wave*16 + mrow is fixed,
//   (a16_n)^q = T_lo[n][q&15] * T_hi[n][q>>4],  q&15 == mrow, q>>4 == wave
// i.e. ONE complex mul + two LDS loads per element (T_lo: contiguous float2
// run across the half-wave; T_hi: half-wave broadcast). Tables are built by
// the 64 setup threads (~24 cmuls each). f32 WMMA has no A/B negate (ISA:
// NEG is C-only for F32), so -S_im is staged instead.

#define HDIM   1024
#define NDIM   64
#define SEQL   2048
#define RDIM   16
#define QDIM   (SEQL / RDIM)   // 128

typedef __attribute__((ext_vector_type(2))) float v2f;
typedef __attribute__((ext_vector_type(8))) float v8f;

__device__ __forceinline__ float2 cmulf(float2 a, float2 b) {
    return make_float2(a.x * b.x - a.y * b.y, a.x * b.y + a.y * b.x);
}

__global__ __launch_bounds__(256) void lssl_vandermonde_wmma(
    const float* __restrict__ Lre, const float* __restrict__ Lim,
    const float* __restrict__ Bre, const float* __restrict__ Bim,
    const float* __restrict__ Cre, const float* __restrict__ Cim,
    const float* __restrict__ Dv,  const float* __restrict__ log_dt,
    float* __restrict__ out)
{
    __shared__ float  sSre [NDIM][RDIM + 1];  //  S_re[n][r]
    __shared__ float  sSimN[NDIM][RDIM + 1];  // -S_im[n][r]
    __shared__ float2 sTlo [NDIM][RDIM + 1];  // a16_n^j,      j = 0..15
    __shared__ float2 sThi [NDIM][9];         // (a16_n^16)^j, j = 0..7

    const int h   = blockIdx.x;
    const int tid = threadIdx.x;

    // ---- per-channel setup (threads 0..63): discretize, build S + pow tables
    if (tid < NDIM) {
        const int n  = tid;
        const int hn = h * NDIM + n;
        const float dt = expf(log_dt[h]);

        float lr = Lre[hn], li = Lim[hn];
        float sp = (lr > 20.0f) ? lr : log1pf(expf(lr));     // softplus
        float lamr = -sp, lami = li;

        float hr = 0.5f * dt * lamr, hi = 0.5f * dt * lami;  // half = 0.5*dt*lam
        float dr = 1.0f - hr, di = -hi;                      // denom = 1 - half
        float nr = 1.0f + hr, ni =  hi;                      // num   = 1 + half
        float inv = 1.0f / (dr * dr + di * di);

        float2 a = make_float2((nr * dr + ni * di) * inv,    // a_bar = num/denom
                               (ni * dr - nr * di) * inv);

        float tr =  dt * dr * inv;                           // dt/denom
        float ti = -dt * di * inv;
        float br = Bre[hn], bi = Bim[hn];
        float2 bb = make_float2(tr * br - ti * bi, tr * bi + ti * br);  // b_bar
        float2 w  = cmulf(make_float2(Cre[hn], Cim[hn]), bb);           // w = C*b_bar

        // S[n][r] = w * a^r, r = 0..15 (store -Im for the WMMA subtract)
        float2 s = w;
        #pragma unroll
        for (int r = 0; r < RDIM; ++r) {
            sSre [n][r] =  s.x;
            sSimN[n][r] = -s.y;
            s = cmulf(s, a);
        }

        // a16 = a^16 via 4 squarings
        float2 a16 = a;
        #pragma unroll
        for (int j = 0; j < 4; ++j) a16 = cmulf(a16, a16);

        // T_lo[n][j] = a16^j (j=0..15); afterwards t == a16^16
        float2 t = make_float2(1.0f, 0.0f);
        #pragma unroll
        for (int j = 0; j < 16; ++j) { sTlo[n][j] = t; t = cmulf(t, a16); }

        // T_hi[n][j] = (a16^16)^j (j=0..7)
        float2 u = make_float2(1.0f, 0.0f);
        #pragma unroll
        for (int j = 0; j < 8; ++j) { sThi[n][j] = u; u = cmulf(u, t); }
    }
    if (tid == 0) out[(size_t)HDIM * SEQL + h] = Dv[h];      // D passthrough
    __syncthreads();

    // ---- WMMA stage: wave w computes K rows q in [16w, 16w+16) ----
    const int wave = tid >> 5;
    const int lane = tid & 31;
    const int mrow = lane & 15;          // A-matrix M index / B-matrix N index
    const int koff = (lane >> 4) << 1;   // lanes 16-31 hold K+2 per ISA layout
    // lane exponent q = wave*16 + mrow:  q & 15 = mrow, q >> 4 = wave

    v8f acc = {};
    #pragma unroll
    for (int kk = 0; kk < 16; ++kk) {
        const int n0 = kk * 4 + koff;    // this lane's two K (=n) indices

        // G[q][n] = T_lo[n][mrow] * T_hi[n][wave]  (one cmul per element)
        float2 g0 = cmulf(sTlo[n0][mrow],     sThi[n0][wave]);
        float2 g1 = cmulf(sTlo[n0 + 1][mrow], sThi[n0 + 1][wave]);

        v2f Are  = { g0.x, g1.x };                     // A 16x4: V0=K0/K2, V1=K1/K3
        v2f Aim  = { g0.y, g1.y };
        v2f Bre_ = { sSre [n0][mrow], sSre [n0 + 1][mrow] };  // B 4x16: V0=row0/2, V1=row1/3
        v2f BimN = { sSimN[n0][mrow], sSimN[n0 + 1][mrow] };

        // acc += G_re @ S_re ; acc += G_im @ (-S_im)
        acc = __builtin_amdgcn_wmma_f32_16x16x4_f32(
                  false, Are, false, Bre_, (short)0, acc, false, false);
        acc = __builtin_amdgcn_wmma_f32_16x16x4_f32(
                  false, Aim, false, BimN, (short)0, acc, false, false);
    }

    // ---- store D-matrix: VGPR v -> M = v (+8 for upper half-wave), N = lane&15
    const int rowBase = wave * 16 + ((lane >> 4) << 3);
    float* outK = out + (size_t)h * SEQL;
    #pragma unroll
    for (int v = 0; v < 8; ++v) {
        outK[(rowBase + v) * RDIM + mrow] = acc[v];
    }
}

extern "C" void kernel_launch(void* const* d_in, const int* in_sizes, int n_in,
                              void* d_out, int out_size, void* d_ws, size_t ws_size,
                              hipStream_t stream) {
    (void)in_sizes; (void)n_in; (void)out_size; (void)d_ws; (void)ws_size;
    const float* Lre    = (const float*)d_in[0];
    const float* Lim    = (const float*)d_in[1];
    const float* Bre    = (const float*)d_in[2];
    const float* Bim    = (const float*)d_in[3];
    const float* Cre    = (const float*)d_in[4];
    const float* Cim    = (const float*)d_in[5];
    const float* Dv     = (const float*)d_in[6];
    const float* log_dt = (const float*)d_in[7];
    float* out = (float*)d_out;

    lssl_vandermonde_wmma<<<dim3(HDIM), dim3(256), 0, stream>>>(
        Lre, Lim, Bre, Bim, Cre, Cim, Dv, log_dt, out);
}